// Affine_Linear_22067541967103
// MI455X (gfx1250) — compile-verified
//
#include <hip/hip_runtime.h>

typedef __attribute__((ext_vector_type(2))) float v2f;
typedef __attribute__((ext_vector_type(8))) float v8f;

#define DD 64          // channel dim (e)
#define FF 64          // output features
#define KK 192         // fused K = 3*DD
#define PTS 16         // points per group (one WMMA column tile)
#define STRIDE 196     // padded LDS row stride for W/T (floats), 64b-bank-conflict-free
#define OSTRIDE 193    // padded LDS row stride for output tile (floats)
#define NTHREADS 128
#define NWAVES 4

__global__ __launch_bounds__(NTHREADS)
void bgs_affine_wmma(const float* __restrict__ X,
                     const float* __restrict__ J,
                     const float* __restrict__ A,
                     const float* __restrict__ Bm,
                     const float* __restrict__ C,
                     float* __restrict__ Y,
                     int ngroups)
{
    __shared__ float ldsW[FF * STRIDE];        // [A | Bm | C], row-major [f][k]
    __shared__ float ldsT[3 * PTS * STRIDE];   // T, layout [(i*16+col)][k]
    __shared__ float ldsO[PTS * OSTRIDE];      // staged output [point][f*3+i]

    const int tid  = threadIdx.x;
    const int wave = tid >> 5;
    const int lane = tid & 31;
    const int ln16 = lane & 15;
    const int lh   = lane >> 4;

    // ---- stage fused weight matrix W = [A | Bm | C] into LDS (once) ----
    #pragma unroll 4
    for (int j = 0; j < (FF * KK) / NTHREADS; ++j) {
        const int idx = tid + j * NTHREADS;
        const int f = idx / KK;
        const int k = idx - f * KK;
        float w;
        if (k < DD)          w = A [f * DD + k];
        else if (k < 2 * DD) w = Bm[f * DD + (k - DD)];
        else                 w = C [f * DD + (k - 2 * DD)];
        ldsW[f * STRIDE + k] = w;
    }
    __syncthreads();

    bool havePrev = false;
    long prev_p0  = 0;

    for (int g = blockIdx.x; g < ngroups; g += gridDim.x) {
        const long p0 = (long)g * PTS;

        // ---- phase 1a: stream out previous group's staged tile (coalesced) ----
        if (havePrev) {
            float* ybase = Y + prev_p0 * (FF * 3);
            #pragma unroll
            for (int j = 0; j < (PTS * FF * 3) / NTHREADS; ++j) {   // 24 iters
                const int idx = tid + j * NTHREADS;
                const int pl  = idx / (FF * 3);
                const int rem = idx - pl * (FF * 3);
                ybase[idx] = ldsO[pl * OSTRIDE + rem];
            }
        }

        // ---- phase 1b: geometry, 16 points x 64 channels = 1024 tasks ----
        #pragma unroll 2
        for (int j = 0; j < (PTS * DD) / NTHREADS; ++j) {           // 8 iters
            const int  task = tid + j * NTHREADS;
            const int  pl = task >> 6;    // local point 0..15
            const int  d  = task & 63;    // channel
            const long pd = p0 * DD + (long)pl * DD + d;
            const float* j6 = J + pd * 6; // J[...,k,i2] -> j6[k*2+i2]
            const float* x3 = X + pd * 3;

            // b1 = normalize(J[:,0])
            float b1x = j6[0], b1y = j6[2], b1z = j6[4];
            float a2x = j6[1], a2y = j6[3], a2z = j6[5];
            const float n1 = sqrtf(b1x*b1x + b1y*b1y + b1z*b1z);
            const float i1 = 1.0f / fmaxf(n1, 1e-12f);
            b1x *= i1; b1y *= i1; b1z *= i1;
            // b2 = normalize(a2 - (b1.a2) b1)
            const float dt = b1x*a2x + b1y*a2y + b1z*a2z;
            const float vx = a2x - dt*b1x, vy = a2y - dt*b1y, vz = a2z - dt*b1z;
            const float n2 = sqrtf(vx*vx + vy*vy + vz*vz);
            const float i2 = 1.0f / fmaxf(n2, 1e-12f);
            const float b2x = vx*i2, b2y = vy*i2, b2z = vz*i2;
            // b3 = b1 x b2
            const float b3x = b1y*b2z - b1z*b2y;
            const float b3y = b1z*b2x - b1x*b2z;
            const float b3z = b1x*b2y - b1y*b2x;
            // RTX = R^T X
            const float x0 = x3[0], x1 = x3[1], x2 = x3[2];
            const float rt0 = b1x*x0 + b1y*x1 + b1z*x2;
            const float rt1 = b2x*x0 + b2y*x1 + b2z*x2;
            const float rt2 = b3x*x0 + b3y*x1 + b3z*x2;
            // terms
            const float ax = b1x*rt0 + b2x*rt1, ay = b1y*rt0 + b2y*rt1, az = b1z*rt0 + b2z*rt1;
            const float bx = b2x*rt0 - b1x*rt1, by = b2y*rt0 - b1y*rt1, bz = b2z*rt0 - b1z*rt1;
            const float cx = b3x*rt2,           cy = b3y*rt2,           cz = b3z*rt2;

            float* t0 = &ldsT[(0 * PTS + pl) * STRIDE];
            float* t1 = &ldsT[(1 * PTS + pl) * STRIDE];
            float* t2 = &ldsT[(2 * PTS + pl) * STRIDE];
            t0[d] = ax; t0[DD + d] = bx; t0[2 * DD + d] = cx;
            t1[d] = ay; t1[DD + d] = by; t1[2 * DD + d] = cy;
            t2[d] = az; t2[DD + d] = bz; t2[2 * DD + d] = cz;
        }

        // ---- phase 1c: prefetch next group's inputs (global_prefetch_b8) ----
        const int gnext = g + gridDim.x;
        if (gnext < ngroups) {
            const long np0 = (long)gnext * PTS * DD;
            const float* jn = J + np0 * 6 + tid * 48;   // 192 B/thread of J
            const float* xn = X + np0 * 3 + tid * 24;   // 96 B/thread of X
            __builtin_prefetch(jn, 0, 3);
            __builtin_prefetch(jn + 32, 0, 3);
            __builtin_prefetch(xn, 0, 3);
        }
        __syncthreads();

        // ---- phase 2: WMMA, 12 tiles of 16x16, K = 192 (48 x 16x16x4 f32) ----
        #pragma unroll
        for (int tj = 0; tj < 12 / NWAVES; ++tj) {      // exactly 3 tiles/wave
            const int t  = wave + tj * NWAVES;
            const int i  = t >> 2;        // output component 0..2
            const int f0 = (t & 3) * 16;  // row tile base
            v8f acc = {};
            const float* wrow = &ldsW[(f0 + ln16) * STRIDE + 2 * lh];
            const float* trow = &ldsT[(i * PTS + ln16) * STRIDE + 2 * lh];
            #pragma unroll 8
            for (int k0 = 0; k0 < KK; k0 += 4) {
                v2f a = *(const v2f*)(wrow + k0);
                v2f b = *(const v2f*)(trow + k0);
                acc = __builtin_amdgcn_wmma_f32_16x16x4_f32(
                        false, a, false, b, (short)0, acc, false, false);
            }
            // D tile -> LDS stage: lane holds N=lane%16 (point), M = r + 8*lh
            float* orow = &ldsO[ln16 * OSTRIDE + i];
            #pragma unroll
            for (int r = 0; r < 8; ++r) {
                const int f = f0 + r + 8 * lh;
                orow[f * 3] = acc[r];
            }
        }
        __syncthreads();

        havePrev = true;
        prev_p0  = p0;
    }

    // ---- epilogue: stream out the final group's staged tile ----
    if (havePrev) {
        float* ybase = Y + prev_p0 * (FF * 3);
        #pragma unroll
        for (int j = 0; j < (PTS * FF * 3) / NTHREADS; ++j) {
            const int idx = tid + j * NTHREADS;
            const int pl  = idx / (FF * 3);
            const int rem = idx - pl * (FF * 3);
            ybase[idx] = ldsO[pl * OSTRIDE + rem];
        }
    }
}

extern "C" void kernel_launch(void* const* d_in, const int* in_sizes, int n_in,
                              void* d_out, int out_size, void* d_ws, size_t ws_size,
                              hipStream_t stream) {
    (void)n_in; (void)out_size; (void)d_ws; (void)ws_size;
    const float* X  = (const float*)d_in[0];
    const float* J  = (const float*)d_in[1];
    const float* A  = (const float*)d_in[2];
    const float* Bm = (const float*)d_in[3];
    const float* C  = (const float*)d_in[4];
    float* Y = (float*)d_out;

    const long npoints = (long)in_sizes[0] / (3L * DD);   // B*N = 65536
    const int  ngroups = (int)(npoints / PTS);            // 4096
    int grid = ngroups < 2048 ? ngroups : 2048;           // persistent groups loop
    if (grid < 1) grid = 1;

    bgs_affine_wmma<<<grid, NTHREADS, 0, stream>>>(X, J, A, Bm, C, Y, ngroups);
}